// ACModel_67559835566374
// MI455X (gfx1250) — compile-verified
//
#include <hip/hip_runtime.h>
#include <cstring>

typedef __attribute__((ext_vector_type(2))) float v2f;
typedef __attribute__((ext_vector_type(8))) float v8f;

// ---------------- workspace layout (float offsets) ----------------
#define WS_OFF_H1    0        // 7*256
#define WS_OFF_H2    1792     // 7*256
#define WS_OFF_BP1   3584     // 16*28*32*2 = 28672  (M1 in B-fragment layout)
#define WS_OFF_BP2   32256    // 28672               (M2 in B-fragment layout)
#define WS_OFF_B1    60928    // 448
#define WS_OFF_B2    61376    // 448
#define WS_OFF_AW1P  61824    // 16*4*32*2 = 4096    (actor_w1 fragments)
#define WS_OFF_CW1P  65920    // 4096                (critic_w1 fragments)
#define WS_OFF_KEY   70016    // 15*64
#define WS_OFF_KVAL  70976    // 16
#define WS_OFF_EEMB  70992    // 15*64

__device__ __forceinline__ v8f wmma_f32_4(v2f a, v2f b, v8f c) {
  // D = A(16x4,f32) * B(4x16,f32) + C(16x16,f32)
  return __builtin_amdgcn_wmma_f32_16x16x4_f32(false, a, false, b, (short)0, c,
                                               false, false);
}

// Four 16x16 output tiles computed simultaneously (K=64 via 16 k-steps of K=4),
// with a manually software-pipelined 4-deep circular buffer of B fragments:
// loads for k-step kk+4 are issued while the WMMAs of kk execute, so ~16
// coalesced 256B loads stay in flight and waits are staggered, not blocking.
__device__ __forceinline__ void wmma_quad(const v2f* a2, const v2f* __restrict__ bp,
                                          int l15, int lh, int lane,
                                          int ntBase, int ntc, v8f acc[4]) {
  v2f buf[4][4];
#pragma unroll
  for (int p = 0; p < 4; ++p)
#pragma unroll
    for (int s = 0; s < 4; ++s)
      buf[p][s] = bp[(p*ntc + ntBase + s)*32 + lane];
#pragma unroll
  for (int kk = 0; kk < 16; ++kk) {
    v2f a  = a2[l15*32 + 2*kk + lh];
    v2f b0 = buf[kk & 3][0], b1 = buf[kk & 3][1];
    v2f b2 = buf[kk & 3][2], b3 = buf[kk & 3][3];
    if (kk + 4 < 16) {
#pragma unroll
      for (int s = 0; s < 4; ++s)
        buf[kk & 3][s] = bp[((kk + 4)*ntc + ntBase + s)*32 + lane];
    }
    acc[0] = wmma_f32_4(a, b0, acc[0]);
    acc[1] = wmma_f32_4(a, b1, acc[1]);
    acc[2] = wmma_f32_4(a, b2, acc[2]);
    acc[3] = wmma_f32_4(a, b3, acc[3]);
  }
}

__device__ __forceinline__ int route_tok(int t) {
  if (t == 1)  return 0;
  if (t == 24) return 1;
  if (t == 5)  return 2;
  if (t == 9)  return 3;
  if (t == 17) return 4;
  if (t == 15) return 5;
  if (t == 26) return 6;
  return 0;
}

// per-thread conv tower (used only for the 15 sampled experience states)
__device__ void conv_embed_st(const float* base, int cs, int ys, int xs,
                              const float* w1, const float* b1,
                              const float* w2, const float* b2,
                              const float* w3, const float* b3, float* out) {
  float pooled[144];
  for (int ch = 0; ch < 16; ++ch)
    for (int py = 0; py < 3; ++py)
      for (int px = 0; px < 3; ++px) {
        float m = 0.f;
        for (int iy = 0; iy < 2; ++iy)
          for (int jx = 0; jx < 2; ++jx) {
            float acc = b1[ch];
            for (int ky = 0; ky < 2; ++ky)
              for (int kx = 0; kx < 2; ++kx)
                for (int c = 0; c < 3; ++c)
                  acc += base[c*cs + (2*py+iy+ky)*ys + (2*px+jx+kx)*xs] *
                         w1[ch*12 + c*4 + ky*2 + kx];
            m = fmaxf(m, acc);
          }
        pooled[ch*9 + py*3 + px] = fmaxf(m, 0.f);
      }
  float o2[128];
  for (int o = 0; o < 32; ++o)
    for (int y = 0; y < 2; ++y)
      for (int x = 0; x < 2; ++x) {
        float acc = b2[o];
        for (int c = 0; c < 16; ++c)
          for (int ky = 0; ky < 2; ++ky)
            for (int kx = 0; kx < 2; ++kx)
              acc += pooled[c*9 + (y+ky)*3 + (x+kx)] * w2[o*64 + c*4 + ky*2 + kx];
        o2[o*4 + y*2 + x] = fmaxf(acc, 0.f);
      }
  for (int o = 0; o < 64; ++o) {
    float acc = b3[o];
    for (int k = 0; k < 128; ++k) acc += o2[k] * w3[o*128 + k];
    out[o] = fmaxf(acc, 0.f);
  }
}

// ---------------- hypernetwork prologue kernels ----------------
__global__ void k_hyper_h(const float* te, const float* ot1, const float* ot2,
                          const float* hw1, const float* hb1, float* ws) {
  int j = threadIdx.x;  // 256
  for (int e = 0; e < 7; ++e) {
    float acc = hb1[j];
    for (int i = 0; i < 128; ++i) acc += te[e*128 + i] * hw1[i*256 + j];
    float a1 = acc, a2 = acc;
    for (int i = 0; i < 8; ++i) {
      float w = hw1[(128+i)*256 + j];
      a1 += ot1[e*8 + i] * w;
      a2 += ot2[e*8 + i] * w;
    }
    ws[WS_OFF_H1 + e*256 + j] = tanhf(a1);
    ws[WS_OFF_H2 + e*256 + j] = tanhf(a2);
  }
}

// M1/M2 directly in WMMA B-fragment layout:
//   Bp[(kk*28+nt)*32 + lane] = { M[e][4kk+2*(lane>>4)][kout], M[e][4kk+2*(lane>>4)+1][kout] }
//   e = nt>>2, kout = (nt&3)*16 + (lane&15)
__global__ void k_hyper_M(const float* hwm, const float* hbm, float* ws) {
  int g = blockIdx.x*blockDim.x + threadIdx.x;
  if (g >= 2*14336) return;
  int m = g / 14336;
  int rem = g % 14336;             // (kk*28+nt)*32+lane
  int lane = rem & 31;
  int ntk = rem >> 5;
  int nt = ntk % 28, kk = ntk / 28;
  int e = nt >> 2;
  int kout = ((nt & 3) << 4) | (lane & 15);
  int d0 = 4*kk + 2*(lane >> 4);
  const float* h = ws + (m ? WS_OFF_H2 : WS_OFF_H1) + e*256;
  int c0 = d0*64 + kout, c1 = c0 + 64;
  float v0 = hbm[c0], v1 = hbm[c1];
  for (int j = 0; j < 256; ++j) {
    float hj = h[j];
    v0 += hj * hwm[j*4096 + c0];
    v1 += hj * hwm[j*4096 + c1];
  }
  float* o = ws + (m ? WS_OFF_BP2 : WS_OFF_BP1);
  o[rem*2] = v0;  o[rem*2+1] = v1;
}

__global__ void k_hyper_b(const float* hwb, const float* hbb, float* ws) {
  int g = blockIdx.x*blockDim.x + threadIdx.x;
  if (g >= 896) return;
  int m = g / 448, rem = g % 448;
  int e = rem / 64, k = rem % 64;
  const float* h = ws + (m ? WS_OFF_H2 : WS_OFF_H1) + e*256;
  float v = hbb[k];
  for (int j = 0; j < 256; ++j) v += h[j] * hwb[j*64 + k];
  ws[(m ? WS_OFF_B2 : WS_OFF_B1) + rem] = v;
}

__global__ void k_packw(const float* aw1, const float* cw1, float* ws) {
  int g = blockIdx.x*blockDim.x + threadIdx.x;
  if (g >= 2*2048) return;
  int m = g / 2048, rem = g % 2048;   // (kk*4+nt)*32+lane
  int lane = rem & 31;
  int ntk = rem >> 5;
  int nt = ntk & 3, kk = ntk >> 2;
  int col = nt*16 + (lane & 15);
  int r0 = 4*kk + 2*(lane >> 4);
  const float* W = m ? cw1 : aw1;
  float* o = ws + (m ? WS_OFF_CW1P : WS_OFF_AW1P);
  o[rem*2]   = W[r0*64 + col];
  o[rem*2+1] = W[(r0+1)*64 + col];
}

// ---------------- experience side path (15 samples) ----------------
struct Samp { int s[15]; };

__global__ void k_exp(const float* es, const int* etext, const float* eval_,
                      const float* w1, const float* b1, const float* w2,
                      const float* b2, const float* w3, const float* b3,
                      Samp samp, float* ws) {
  __shared__ int kidx[15];
  int t = threadIdx.x;  // 960
  if (t < 15) {
    int s = samp.s[t];
    conv_embed_st(es + s*147, 49, 7, 1, w1, b1, w2, b2, w3, b3,
                  ws + WS_OFF_EEMB + t*64);
    kidx[t] = route_tok(etext[s*4]);
    ws[WS_OFF_KVAL + t] = eval_[s];
  }
  __syncthreads();
  int i = t / 64, k = t % 64;
  int e = kidx[i];
  float acc = ws[WS_OFF_B2 + e*64 + k];
  const float* bp2 = ws + WS_OFF_BP2;
  const float* ee = ws + WS_OFF_EEMB + i*64;
  int nt = e*4 + (k >> 4);
  for (int d = 0; d < 64; ++d) {
    int kk = d >> 2;
    int lane = (((d >> 1) & 1) << 4) | (k & 15);
    acc += ee[d] * bp2[((kk*28 + nt)*32 + lane)*2 + (d & 1)];
  }
  ws[WS_OFF_KEY + i*64 + k] = acc;
}

// ---------------- main fused kernel ----------------
// 8 waves/block, 2 reps/wave, 16 rows/tile -> 256 rows/block.
// Per-wave LDS arena:
//   region A [0,2352): img  -> o2 -> {emb_hyper[0..1023], actorH[1024..2047]}
//   region B [2352,4656): pooled -> {emb[0..1023], criticH[1024..2047]}
#define ARENA 4656

__global__ __launch_bounds__(256) void k_main(
    const float* __restrict__ img, const int* __restrict__ text,
    const float* __restrict__ ws,
    const float* __restrict__ w1, const float* __restrict__ b1c,
    const float* __restrict__ w2, const float* __restrict__ b2c,
    const float* __restrict__ w3, const float* __restrict__ b3c,
    const float* __restrict__ ab1, const float* __restrict__ aw2,
    const float* __restrict__ ab2, const float* __restrict__ cb1,
    const float* __restrict__ cw2, const float* __restrict__ cb2,
    float* __restrict__ out, int B) {
  __shared__ float s_arena[8 * ARENA];
  __shared__ float s_w1[192], s_b1c[16], s_w2[2048], s_b2c[32];
  __shared__ float s_w3[8192], s_b3c[64];
  __shared__ float s_keys[960], s_kval[16], s_b1[448], s_b2[448];
  __shared__ int   s_idxw[128];

  const int tid = threadIdx.x;
  for (int i = tid; i < 192;  i += 256) s_w1[i] = w1[i];
  if (tid < 16) s_b1c[tid] = b1c[tid];
  for (int i = tid; i < 2048; i += 256) s_w2[i] = w2[i];
  if (tid < 32) s_b2c[tid] = b2c[tid];
  for (int i = tid; i < 8192; i += 256) s_w3[i] = w3[i];
  if (tid < 64) s_b3c[tid] = b3c[tid];
  for (int i = tid; i < 960;  i += 256) s_keys[i] = ws[WS_OFF_KEY + i];
  if (tid < 15) s_kval[tid] = ws[WS_OFF_KVAL + tid];
  for (int i = tid; i < 448;  i += 256) {
    s_b1[i] = ws[WS_OFF_B1 + i];
    s_b2[i] = ws[WS_OFF_B2 + i];
  }
  __syncthreads();

  const int wv = tid >> 5, lane = tid & 31;
  const int l15 = lane & 15, lh = lane >> 4;
  const v2f* bp1 = (const v2f*)(ws + WS_OFF_BP1);
  const v2f* bp2 = (const v2f*)(ws + WS_OFF_BP2);
  const v2f* awp = (const v2f*)(ws + WS_OFF_AW1P);
  const v2f* cwp = (const v2f*)(ws + WS_OFF_CW1P);
  float* aimg  = &s_arena[wv * ARENA];
  float* apool = aimg + 2352;

#pragma unroll 1
  for (int rep = 0; rep < 2; ++rep) {
    const int tileR = (wv*2 + rep) * 16;             // block-local row base
    const int growBase = blockIdx.x*256 + tileR;     // global row base

    // ---- coalesced image tile load: 16 rows * 147 floats ----
    const float* gimg = img + (size_t)growBase * 147;
    for (int i = lane; i < 2352; i += 32) aimg[i] = gimg[i];
    if (lane < 16) s_idxw[wv*16 + lane] = route_tok(text[(growBase + lane)*4]);
    if (rep == 0) {  // prefetch next rep's image tile into caches
      const char* nx = (const char*)(img + (size_t)(growBase + 16) * 147);
      __builtin_prefetch(nx + lane*128, 0, 0);
      __builtin_prefetch(nx + 4096 + lane*128, 0, 0);
      __builtin_prefetch(nx + 8192 + lane*128, 0, 0);
    }

    // ---- conv1 + relu + maxpool (lane-cooperative) ----
#pragma unroll 1
    for (int t = 0; t < 72; ++t) {
      int id = lane + 32*t;                 // (r, ch, pos) 16*16*9
      int r = id / 144, rem = id % 144;
      int ch = rem / 9, pos = rem % 9;
      int py = pos / 3, px = pos % 3;
      const float* ib = aimg + r*147;
      float m = 0.f;
      for (int iy = 0; iy < 2; ++iy)
        for (int jx = 0; jx < 2; ++jx) {
          float acc = s_b1c[ch];
#pragma unroll
          for (int ky = 0; ky < 2; ++ky)
#pragma unroll
            for (int kx = 0; kx < 2; ++kx)
#pragma unroll
              for (int c = 0; c < 3; ++c)
                acc += ib[(2*py+iy+ky)*21 + (2*px+jx+kx)*3 + c] *
                       s_w1[ch*12 + c*4 + ky*2 + kx];
          m = fmaxf(m, acc);
        }
      apool[id] = fmaxf(m, 0.f);            // [r][ch][pos]
    }
    // ---- conv2 + relu (o2 overlays img region) ----
    float* ao2 = aimg;
#pragma unroll 1
    for (int t = 0; t < 64; ++t) {
      int id = lane + 32*t;                 // (r, o, pos) 16*32*4
      int r = id >> 7, rem = id & 127;
      int o = rem >> 2, pos = rem & 3;
      int y = pos >> 1, x = pos & 1;
      const float* pb = apool + r*144;
      float acc = s_b2c[o];
      for (int c = 0; c < 16; ++c)
#pragma unroll
        for (int ky = 0; ky < 2; ++ky)
#pragma unroll
          for (int kx = 0; kx < 2; ++kx)
            acc += pb[c*9 + (y+ky)*3 + (x+kx)] * s_w2[o*64 + c*4 + ky*2 + kx];
      ao2[id] = fmaxf(acc, 0.f);            // [r][o*4+pos]
    }
    // ---- conv3 + relu -> emb (overlays pooled region) ----
    float* aemb = apool;
#pragma unroll 1
    for (int t = 0; t < 32; ++t) {
      int id = lane + 32*t;                 // (r, o) 16*64
      int r = id >> 6, o = id & 63;
      const float* ob = ao2 + r*128;
      float acc = s_b3c[o];
      for (int k = 0; k < 128; ++k) acc += ob[k] * s_w3[o*128 + k];
      aemb[id] = fmaxf(acc, 0.f);
    }

    int idxv[8];
#pragma unroll
    for (int v = 0; v < 8; ++v) idxv[v] = s_idxw[wv*16 + v + 8*lh];

    float* stg = aimg;  // region A: emb_hyper [0..1023], actorH [1024..2047]

    // ---- expert GEMM #1: emb @ M1cat(64x448), select expert per row ----
    {
      v8f hyp[4];
#pragma unroll
      for (int s = 0; s < 4; ++s) hyp[s] = (v8f){0, 0, 0, 0, 0, 0, 0, 0};
#pragma unroll 1
      for (int e = 0; e < 7; ++e) {
        v8f acc[4];
#pragma unroll
        for (int s = 0; s < 4; ++s) acc[s] = (v8f){0, 0, 0, 0, 0, 0, 0, 0};
        wmma_quad((const v2f*)aemb, bp1, l15, lh, lane, e*4, 28, acc);
        bool take[8];
#pragma unroll
        for (int v = 0; v < 8; ++v) take[v] = (idxv[v] == e);
#pragma unroll
        for (int s = 0; s < 4; ++s)
#pragma unroll
          for (int v = 0; v < 8; ++v)
            hyp[s][v] = take[v] ? acc[s][v] : hyp[s][v];
      }
#pragma unroll
      for (int s = 0; s < 4; ++s)
#pragma unroll
        for (int v = 0; v < 8; ++v)
          stg[(v + 8*lh)*64 + s*16 + l15] =
              hyp[s][v] + s_b1[idxv[v]*64 + s*16 + l15];
    }

    // ---- actor hidden: tanh(emb_hyper @ W + b) -> stg[1024..2047] ----
    {
      v8f acc[4];
#pragma unroll
      for (int s = 0; s < 4; ++s) acc[s] = (v8f){0, 0, 0, 0, 0, 0, 0, 0};
      wmma_quad((const v2f*)stg, awp, l15, lh, lane, 0, 4, acc);
#pragma unroll
      for (int s = 0; s < 4; ++s) {
        float bias = ab1[s*16 + l15];
#pragma unroll
        for (int v = 0; v < 8; ++v)
          stg[1024 + (v + 8*lh)*64 + s*16 + l15] = tanhf(acc[s][v] + bias);
      }
    }
    // ---- critic hidden -> apool[1024..2047] ----
    {
      v8f acc[4];
#pragma unroll
      for (int s = 0; s < 4; ++s) acc[s] = (v8f){0, 0, 0, 0, 0, 0, 0, 0};
      wmma_quad((const v2f*)stg, cwp, l15, lh, lane, 0, 4, acc);
#pragma unroll
      for (int s = 0; s < 4; ++s) {
        float bias = cb1[s*16 + l15];
#pragma unroll
        for (int v = 0; v < 8; ++v)
          apool[1024 + (v + 8*lh)*64 + s*16 + l15] = tanhf(acc[s][v] + bias);
      }
    }
    // ---- heads: log_softmax(logits) / value ----
    if (lane < 16) {
      int gr = growBase + lane;
      const float* h = &stg[1024 + lane*64];
      float lg[7]; float mx = -3.4e38f;
#pragma unroll
      for (int a = 0; a < 7; ++a) {
        float acc = ab2[a];
        for (int d = 0; d < 64; ++d) acc += h[d] * aw2[d*7 + a];
        lg[a] = acc; mx = fmaxf(mx, acc);
      }
      float Z = 0.f;
#pragma unroll
      for (int a = 0; a < 7; ++a) Z += expf(lg[a] - mx);
      float lse = logf(Z) + mx;
#pragma unroll
      for (int a = 0; a < 7; ++a) out[(size_t)gr*7 + a] = lg[a] - lse;
    } else {
      int gr = growBase + (lane - 16);
      const float* h = &apool[1024 + (lane - 16)*64];
      float acc = cb2[0];
      for (int d = 0; d < 64; ++d) acc += h[d] * cw2[d];
      out[(size_t)B*7 + gr] = acc;
    }

    // ---- expert GEMM #2: query = select(emb @ M2cat) + b2[idx] ----
    {
      v8f q[4];
#pragma unroll
      for (int s = 0; s < 4; ++s) q[s] = (v8f){0, 0, 0, 0, 0, 0, 0, 0};
#pragma unroll 1
      for (int e = 0; e < 7; ++e) {
        v8f acc[4];
#pragma unroll
        for (int s = 0; s < 4; ++s) acc[s] = (v8f){0, 0, 0, 0, 0, 0, 0, 0};
        wmma_quad((const v2f*)aemb, bp2, l15, lh, lane, e*4, 28, acc);
        bool take[8];
#pragma unroll
        for (int v = 0; v < 8; ++v) take[v] = (idxv[v] == e);
#pragma unroll
        for (int s = 0; s < 4; ++s)
#pragma unroll
          for (int v = 0; v < 8; ++v)
            q[s][v] = take[v] ? acc[s][v] : q[s][v];
      }
#pragma unroll
      for (int s = 0; s < 4; ++s)
#pragma unroll
        for (int v = 0; v < 8; ++v)
          stg[(v + 8*lh)*64 + s*16 + l15] =
              q[s][v] + s_b2[idxv[v]*64 + s*16 + l15];
    }

    // ---- attention head: softmax(query @ key^T) -> lam, memory_value ----
    if (lane < 16) {
      int gr = growBase + lane;
      const float* qv = &stg[lane*64];
      float sarr[15]; float mx = -3.4e38f;
#pragma unroll
      for (int i = 0; i < 15; ++i) {
        float s = 0.f;
        for (int d = 0; d < 64; ++d) s += qv[d] * s_keys[i*64 + d];
        sarr[i] = s; mx = fmaxf(mx, s);
      }
      float Z = 0.f, mv = 0.f;
#pragma unroll
      for (int i = 0; i < 15; ++i) {
        float p = expf(sarr[i] - mx);
        Z += p; mv += p * s_kval[i];
      }
      out[(size_t)B*9 + gr] = 1.f / Z;   // lam = max softmax
      out[(size_t)B*8 + gr] = mv / Z;    // memory_value
    }
  }
}

__global__ void k_copy(const float4* __restrict__ src, float4* __restrict__ dst,
                       int n4) {
  int i = blockIdx.x*blockDim.x + threadIdx.x;
  if (i < n4) dst[i] = src[i];
}

// ---------------- host: numpy RandomState(0).permutation(50)[:15] ----------------
static void mt_sample15(int* out) {
  unsigned mt[624];
  mt[0] = 0u;
  for (int i = 1; i < 624; ++i)
    mt[i] = 1812433253u * (mt[i-1] ^ (mt[i-1] >> 30)) + (unsigned)i;
  int mti = 624;
  auto genrand = [&]() -> unsigned {
    if (mti >= 624) {
      for (int k = 0; k < 624; ++k) {
        unsigned y = (mt[k] & 0x80000000u) | (mt[(k+1) % 624] & 0x7fffffffu);
        mt[k] = mt[(k+397) % 624] ^ (y >> 1) ^ ((y & 1u) ? 2567483615u : 0u);
      }
      mti = 0;
    }
    unsigned y = mt[mti++];
    y ^= y >> 11; y ^= (y << 7) & 2636928640u;
    y ^= (y << 15) & 4022730752u; y ^= y >> 18;
    return y;
  };
  int arr[50];
  for (int i = 0; i < 50; ++i) arr[i] = i;
  for (int i = 49; i >= 1; --i) {
    unsigned mx = (unsigned)i, mask = mx;
    mask |= mask >> 1; mask |= mask >> 2; mask |= mask >> 4;
    mask |= mask >> 8; mask |= mask >> 16;
    unsigned v;
    do { v = genrand() & mask; } while (v > mx);
    int j = (int)v, t = arr[i]; arr[i] = arr[j]; arr[j] = t;
  }
  for (int i = 0; i < 15; ++i) out[i] = arr[i];
}

extern "C" void kernel_launch(void* const* d_in, const int* in_sizes, int n_in,
                              void* d_out, int out_size, void* d_ws, size_t ws_size,
                              hipStream_t stream) {
  const float* img   = (const float*)d_in[0];
  const int*   text  = (const int*)  d_in[1];
  const float* mem   = (const float*)d_in[2];
  const float* es    = (const float*)d_in[4];
  const float* evalv = (const float*)d_in[5];
  const int*   etext = (const int*)  d_in[6];
  const float* c1w = (const float*)d_in[7],  *c1b = (const float*)d_in[8];
  const float* c2w = (const float*)d_in[9],  *c2b = (const float*)d_in[10];
  const float* c3w = (const float*)d_in[11], *c3b = (const float*)d_in[12];
  const float* aw1 = (const float*)d_in[13], *ab1 = (const float*)d_in[14];
  const float* aw2 = (const float*)d_in[15], *ab2 = (const float*)d_in[16];
  const float* cw1 = (const float*)d_in[17], *cb1 = (const float*)d_in[18];
  const float* cw2 = (const float*)d_in[19], *cb2 = (const float*)d_in[20];
  const float* hw1 = (const float*)d_in[21], *hb1 = (const float*)d_in[22];
  const float* hwm = (const float*)d_in[23], *hbm = (const float*)d_in[24];
  const float* hwb = (const float*)d_in[25], *hbb = (const float*)d_in[26];
  const float* temb = (const float*)d_in[27];
  const float* ot1  = (const float*)d_in[28];
  const float* ot2  = (const float*)d_in[29];
  float* ws  = (float*)d_ws;
  float* out = (float*)d_out;
  const int B = in_sizes[0] / 147;

  Samp samp;
  mt_sample15(samp.s);

  k_hyper_h<<<1, 256, 0, stream>>>(temb, ot1, ot2, hw1, hb1, ws);
  k_hyper_M<<<112, 256, 0, stream>>>(hwm, hbm, ws);
  k_hyper_b<<<4, 256, 0, stream>>>(hwb, hbb, ws);
  k_packw<<<16, 256, 0, stream>>>(aw1, cw1, ws);
  k_exp<<<1, 960, 0, stream>>>(es, etext, evalv, c1w, c1b, c2w, c2b, c3w, c3b,
                               samp, ws);
  k_main<<<B/256, 256, 0, stream>>>(img, text, ws, c1w, c1b, c2w, c2b, c3w, c3b,
                                    ab1, aw2, ab2, cb1, cw2, cb2, out, B);
  int n4 = (B * 128) / 4;
  k_copy<<<(n4 + 255)/256, 256, 0, stream>>>((const float4*)mem,
                                             (float4*)(out + (size_t)B*10), n4);
}